// KStepRGCN_429496729897
// MI455X (gfx1250) — compile-verified
//
#include <hip/hip_runtime.h>

// RGCN (basis decomposition), K=3 layers, N=50000 nodes, E=600000 edges,
// D=128, R=3 relations, B=3 bases. FP32 throughout using V_WMMA_F32_16X16X4_F32.

#define DIMD 128
#define RREL 3
#define BBAS 3
#define KLAY 3
#define MT   8      // m-tiles per wave in the GEMM (B-panel register reuse factor)

typedef __attribute__((ext_vector_type(2))) float v2f;
typedef __attribute__((ext_vector_type(8))) float v8f;

// ---------------------------------------------------------------------------
// W[r] = sum_b att[r,b] * basis[b]   (R*D*D elements, trivial)
// ---------------------------------------------------------------------------
__global__ void basis_combine_kernel(const float* __restrict__ att,    // [R,B]
                                     const float* __restrict__ basis,  // [B,D,D]
                                     float* __restrict__ W)            // [R,D,D]
{
    int idx = blockIdx.x * blockDim.x + threadIdx.x;
    if (idx >= RREL * DIMD * DIMD) return;
    int r  = idx / (DIMD * DIMD);
    int od = idx % (DIMD * DIMD);
    float acc = 0.f;
#pragma unroll
    for (int b = 0; b < BBAS; ++b)
        acc += att[r * BBAS + b] * basis[(size_t)b * DIMD * DIMD + od];
    W[idx] = acc;
}

// ---------------------------------------------------------------------------
// GEMM: out_j = H @ Wj  for j in {0,1,2 (relations), 3 (root)}.
//
// One wave32 owns one 16-column strip of one weight matrix. It hoists the
// whole 128x16 B panel into a register array (v2f bpan[32], 64 VGPRs), then
// sweeps MT m-tiles: each 16x16 output tile is 32x V_WMMA_F32_16X16X4_F32
// fed by a single global_load_b64 of A per WMMA (B comes from registers).
//
// grid: (ceil(N/16 / (4*MT)), D/16, 4), block = 128 threads (4 waves).
//
// VGPR layouts per CDNA5 ISA 7.12.2:
//   A (16x4 fp32):  lane<16 -> M=lane, v0=K0 v1=K1 ; lane>=16 -> K2/K3
//   B (4x16 fp32):  lane%16 = N,       v0=K0(K2)  v1=K1(K3) per half-wave
//   C/D (16x16):    vgpr j : lanes0-15 row m0+j, lanes16-31 row m0+j+8
// ---------------------------------------------------------------------------
__global__ void __launch_bounds__(128)
rgcn_gemm_kernel(const float* __restrict__ H,      // [N,D]
                 const float* __restrict__ W3,     // [R,D,D]
                 const float* __restrict__ root,   // [D,D]
                 float* __restrict__ xw,           // [R,N,D]
                 float* __restrict__ hroot,        // [N,D]
                 int N)
{
    const int lane = threadIdx.x & 31;
    const int wave = threadIdx.x >> 5;
    const int n0   = blockIdx.y << 4;
    const int j    = blockIdx.z;

    const float* Wm  = (j < RREL) ? (W3 + (size_t)j * DIMD * DIMD) : root;
    float*       out = (j < RREL) ? (xw + (size_t)j * N * DIMD)    : hroot;

    const int col  = n0 + (lane & 15);          // B/C column this lane feeds
    const int koff = (lane >> 4) << 1;          // half-wave K offset (0 or 2)

    // ---- hoist the 128x16 B panel into registers (reused for MT m-tiles) ---
    const float* bcol = Wm + (size_t)koff * DIMD + col;
    v2f bpan[32];
#pragma unroll
    for (int t = 0; t < 32; ++t) {
        bpan[t].x = bcol[(size_t)(4 * t) * DIMD];          // Wm[4t+koff  ][col]
        bpan[t].y = bcol[(size_t)(4 * t) * DIMD + DIMD];   // Wm[4t+koff+1][col]
    }

    // ---- sweep MT m-tiles with this B panel -------------------------------
    const int tile0 = (blockIdx.x * 4 + wave) * MT;
    for (int mt = 0; mt < MT; ++mt) {
        const int m0 = (tile0 + mt) << 4;
        if (m0 >= N) break;                      // wave-uniform: EXEC stays all-1s

        const float* arow = H + (size_t)(m0 + (lane & 15)) * DIMD + koff;
        v8f c = {};
#pragma unroll
        for (int t = 0; t < 32; ++t) {
            v2f a = *(const v2f*)(arow + 4 * t); // H[row][4t+koff .. +1] (b64)
            c = __builtin_amdgcn_wmma_f32_16x16x4_f32(
                    /*neg_a=*/false, a, /*neg_b=*/false, bpan[t],
                    /*c_mod=*/(short)0, c, /*reuse_a=*/false, /*reuse_b=*/false);
        }

        const int rbase = m0 + ((lane >> 4) << 3);
#pragma unroll
        for (int jj = 0; jj < 8; ++jj)
            out[(size_t)(rbase + jj) * DIMD + col] = c[jj];
    }
}

// ---------------------------------------------------------------------------
// Zero the aggregation buffers (harness does not re-zero ws between replays).
// ---------------------------------------------------------------------------
__global__ void zero_kernel(float* __restrict__ aggr, float* __restrict__ cnt,
                            int n_aggr, int n_cnt)
{
    int i = blockIdx.x * blockDim.x + threadIdx.x;
    if (i < n_aggr) aggr[i] = 0.f;
    if (i < n_cnt)  cnt[i]  = 0.f;
}

// ---------------------------------------------------------------------------
// Edge phase: one wave32 per edge. Lane l moves 4 contiguous floats
// (float4 -> global_load_b128) of xw[rel, src] and atomically adds them into
// aggr[dst]. xw (76.8MB) and aggr (25.6MB) fit in the 192MB L2, so the random
// gather/scatter is L2-resident. Lane 0 bumps the degree counter.
// ---------------------------------------------------------------------------
__global__ void __launch_bounds__(256)
rgcn_msg_kernel(const float* __restrict__ xw,   // [R,N,D]
                const int* __restrict__ src,
                const int* __restrict__ dst,
                const int* __restrict__ rel,
                float* __restrict__ aggr,       // [N,D]
                float* __restrict__ cnt,        // [N]
                int E, int N)
{
    const int e = blockIdx.x * (blockDim.x >> 5) + (threadIdx.x >> 5);
    if (e >= E) return;
    const int lane = threadIdx.x & 31;
    const int s = src[e];
    const int d = dst[e];
    const int r = rel[e];

    const float4 v = *(const float4*)(xw + ((size_t)r * N + s) * DIMD + (lane << 2));
    float* o = aggr + (size_t)d * DIMD + (lane << 2);
    atomicAdd(o + 0, v.x);
    atomicAdd(o + 1, v.y);
    atomicAdd(o + 2, v.z);
    atomicAdd(o + 3, v.w);
    if (lane == 0) atomicAdd(cnt + d, 1.0f);
}

// ---------------------------------------------------------------------------
// out = aggr / max(cnt,1) + H@root + bias  (+ PReLU except last layer)
// ---------------------------------------------------------------------------
__global__ void rgcn_combine_kernel(const float* __restrict__ aggr,
                                    const float* __restrict__ cnt,
                                    const float* __restrict__ hroot,
                                    const float* __restrict__ bias,     // [D]
                                    const float* __restrict__ prelu_a,  // [1]
                                    float* __restrict__ out,
                                    int N, int act)
{
    int i = blockIdx.x * blockDim.x + threadIdx.x;
    if (i >= N * DIMD) return;
    int row = i >> 7;      // / 128
    int col = i & 127;
    float dnm = cnt[row];
    dnm = dnm > 1.0f ? dnm : 1.0f;
    float v = aggr[i] / dnm + hroot[i] + bias[col];
    if (act) {
        float a = prelu_a[0];
        v = (v >= 0.f) ? v : a * v;
    }
    out[i] = v;
}

// ---------------------------------------------------------------------------
// Orchestration: 3 layers, all on `stream`, scratch carved out of d_ws.
// ---------------------------------------------------------------------------
extern "C" void kernel_launch(void* const* d_in, const int* in_sizes, int n_in,
                              void* d_out, int out_size, void* d_ws, size_t ws_size,
                              hipStream_t stream)
{
    const float* x     = (const float*)d_in[0];   // [N,D]
    const int*   ei    = (const int*)  d_in[1];   // [2,E]
    const int*   ea    = (const int*)  d_in[2];   // [E]
    const float* basis = (const float*)d_in[3];   // [K,B,D,D]
    const float* att   = (const float*)d_in[4];   // [K,R,B]
    const float* root  = (const float*)d_in[5];   // [K,D,D]
    const float* bias  = (const float*)d_in[6];   // [K,D]
    const float* prelu = (const float*)d_in[7];   // [1]

    const int N = in_sizes[0] / DIMD;
    const int E = in_sizes[2];
    const int* src = ei;
    const int* dst = ei + E;

    const size_t ND = (size_t)N * DIMD;
    float* ws    = (float*)d_ws;
    float* Hbuf  = ws;  ws += ND;                    // inter-layer activations
    float* xw    = ws;  ws += (size_t)RREL * ND;     // [R,N,D]
    float* hroot = ws;  ws += ND;                    // H @ root
    float* aggr  = ws;  ws += ND;                    // scatter accumulator
    float* cntb  = ws;  ws += N;                     // degree counts
    float* W3    = ws;  ws += (size_t)RREL * DIMD * DIMD;

    const int mTiles = (N + 15) / 16;                         // 3125
    const int gemmX  = (mTiles + 4 * MT - 1) / (4 * MT);      // m-chunks

    const dim3 bB(256), gB((RREL * DIMD * DIMD + 255) / 256);
    const dim3 bG(128), gG(gemmX, DIMD / 16, RREL + 1);
    const dim3 bM(256), gM((E + 7) / 8);            // 8 edges (waves) per block
    const dim3 bZ(256), gZ((unsigned)((ND + 255) / 256));
    const dim3 bC(256), gC((unsigned)((ND + 255) / 256));

    for (int k = 0; k < KLAY; ++k) {
        basis_combine_kernel<<<gB, bB, 0, stream>>>(
            att + (size_t)k * RREL * BBAS,
            basis + (size_t)k * BBAS * DIMD * DIMD, W3);

        const float* Hin = (k == 0) ? x : Hbuf;
        rgcn_gemm_kernel<<<gG, bG, 0, stream>>>(
            Hin, W3, root + (size_t)k * DIMD * DIMD, xw, hroot, N);

        zero_kernel<<<gZ, bZ, 0, stream>>>(aggr, cntb, (int)ND, N);

        rgcn_msg_kernel<<<gM, bM, 0, stream>>>(xw, src, dst, ea, aggr, cntb, E, N);

        float* outp = (k == KLAY - 1) ? (float*)d_out : Hbuf;
        rgcn_combine_kernel<<<gC, bC, 0, stream>>>(
            aggr, cntb, hroot, bias + (size_t)k * DIMD, prelu,
            outp, N, (k != KLAY - 1));
    }
}